// GeometricWarper_10625749090356
// MI455X (gfx1250) — compile-verified
//
#include <hip/hip_runtime.h>
#include <hip/hip_bf16.h>
#include <stdint.h>

// Problem constants (from the reference)
constexpr int Bn    = 16;
constexpr int Tn    = 480000;
constexpr int CHUNK = 400;
constexpr int Kn    = Tn / CHUNK;   // 1200
constexpr int ROWS  = Bn * 2;       // 32 (b, ear) rows

// Tiling for the main gather kernel
constexpr int TILE = 8192;          // output samples per block
constexpr int HALO = 2048;          // backward reach covered by LDS tile
constexpr int LDSF = HALO + TILE + 16;

// ---------------------------------------------------------------------------
// CDNA5 async global->LDS helpers (ASYNCcnt path, gfx1250-specific)
// ---------------------------------------------------------------------------
typedef int   v4i __attribute__((vector_size(16)));
typedef float v4f __attribute__((ext_vector_type(4)));
typedef __attribute__((address_space(1))) v4i* g_v4i_ptr;   // global
typedef __attribute__((address_space(3))) v4i* l_v4i_ptr;   // LDS

__device__ __forceinline__ void async_g2l_b128(const float* g, float* l) {
#if __has_builtin(__builtin_amdgcn_global_load_async_to_lds_b128)
  __builtin_amdgcn_global_load_async_to_lds_b128(
      (g_v4i_ptr)g, (l_v4i_ptr)l, 0, 0);
#else
  unsigned lds_off = (unsigned)(uintptr_t)(__attribute__((address_space(3))) float*)l;
  unsigned long long ga = (unsigned long long)(uintptr_t)g;
  asm volatile("global_load_async_to_lds_b128 %0, %1, off"
               :: "v"(lds_off), "v"(ga) : "memory");
#endif
}

__device__ __forceinline__ void wait_async_zero() {
#if __has_builtin(__builtin_amdgcn_s_wait_asynccnt)
  __builtin_amdgcn_s_wait_asynccnt(0);
#else
  asm volatile("s_wait_asynccnt 0" ::: "memory");
#endif
}

// ---------------------------------------------------------------------------
// Kernel 1: per (b,k) quaternion -> warp field  w = (-dist/343)*48000
// warp layout: [(b*2+ear)*Kn + k]
// ---------------------------------------------------------------------------
__global__ void gw_disp_kernel(const float* __restrict__ view,
                               float* __restrict__ warp) {
  int id = blockIdx.x * blockDim.x + threadIdx.x;
  if (id >= Bn * Kn) return;
  int b = id / Kn;
  int k = id - b * Kn;
  const float* vb = view + (size_t)b * 7 * Kn + k;

  float px = vb[0 * Kn], py = vb[1 * Kn], pz = vb[2 * Kn];
  float qx = vb[3 * Kn], qy = vb[4 * Kn], qz = vb[5 * Kn], qw = vb[6 * Kn];

  float n2 = qx * qx + qy * qy + qz * qz + qw * qw;
  if (n2 == 0.0f) { qx = 1.0f; qy = 1.0f; qz = 1.0f; qw = 1.0f; n2 = 4.0f; }
  float inv = 1.0f / sqrtf(n2);
  qx *= inv; qy *= inv; qz *= inv; qw *= inv;

  float xx = qx * qx, yy = qy * qy, zz = qz * qz;
  float xy = qx * qy, xz = qx * qz, yz = qy * qz;
  float xw = qx * qw, yw = qy * qw, zw = qz * qw;

  // mouth_i = sum_j R[j][i] * M[j],  M = (0.09, 0, -0.2)   (einsum 'nji,j->ni')
  float m0 = 0.09f * (1.0f - 2.0f * (yy + zz)) + (-0.2f) * (2.0f * (xz - yw));
  float m1 = 0.09f * (2.0f * (xy - zw))        + (-0.2f) * (2.0f * (yz + xw));
  float m2 = 0.09f * (2.0f * (xz + yw))        + (-0.2f) * (1.0f - 2.0f * (xx + yy));

  float sx = px + m0, sy = py + m1, sz = pz + m2;
  // L ear (0,-0.08,-0.22), R ear (0,0.08,-0.22)
  float dlx = sx, dly = sy + 0.08f, dlz = sz + 0.22f;
  float drx = sx, dry = sy - 0.08f, drz = sz + 0.22f;

  float dl = sqrtf(dlx * dlx + dly * dly + dlz * dlz);
  float dr = sqrtf(drx * drx + dry * dry + drz * drz);

  warp[((size_t)b * 2 + 0) * Kn + k] = (-dl / 343.0f) * 48000.0f;
  warp[((size_t)b * 2 + 1) * Kn + k] = (-dr / 343.0f) * 48000.0f;
}

// ---------------------------------------------------------------------------
// Kernel 2: per-row exclusive prefix-max over chunk-end values.
//   e_k = clip(w_k + (400k+399), 0, T-1);  pprev[k] = max(e_0..e_{k-1}) (0 for k=0)
// Also computes reach[row] = max_k(-w_k) for the LDS-halo guarantee.
// One wave32 per row; 1200-long scan via shuffle scans.
// ---------------------------------------------------------------------------
__global__ void gw_scan_kernel(const float* __restrict__ warp,
                               float* __restrict__ pprev,
                               float* __restrict__ reach) {
  int row  = blockIdx.x;
  int lane = threadIdx.x;           // 0..31 (wave32)
  const float* w = warp + (size_t)row * Kn;

  float carry  = 0.0f;              // pos >= 0 always, so 0 is the identity
  float mreach = 0.0f;

  for (int base = 0; base < Kn; base += 32) {
    int k = base + lane;
    float wv = (k < Kn) ? w[k] : 0.0f;
    float e  = (k < Kn)
        ? fminf(fmaxf(wv + (float)(k * CHUNK + (CHUNK - 1)), 0.0f), (float)(Tn - 1))
        : -1.0f;                    // inert for invalid lanes (valid e >= 0)
    if (k < Kn) mreach = fmaxf(mreach, -wv);

    // wave32 inclusive max-scan
    float x = e;
    #pragma unroll
    for (int off = 1; off < 32; off <<= 1) {
      float y = __shfl_up(x, off, 32);
      if (lane >= off) x = fmaxf(x, y);
    }
    float excl = __shfl_up(x, 1, 32);
    float p = (lane == 0) ? carry : fmaxf(carry, excl);
    if (k < Kn) pprev[(size_t)row * Kn + k] = p;
    carry = fmaxf(carry, __shfl(x, 31, 32));
  }

  #pragma unroll
  for (int off = 16; off >= 1; off >>= 1)
    mreach = fmaxf(mreach, __shfl_xor(mreach, off, 32));
  if (lane == 0) reach[row] = mreach;
}

// ---------------------------------------------------------------------------
// Kernel 3: main gather. One block = one (row, 8192-sample tile).
// Stages mono[tileStart-HALO, tileStart+TILE) into LDS with async B128 copies,
// then:  p = max(clip(w_k + t, 0, T-1), pprev_k);  lerp gather from LDS.
// Each lane produces 4 consecutive samples -> one 16B non-temporal store.
// ---------------------------------------------------------------------------
__global__ __launch_bounds__(256)
void gw_warp_kernel(const float* __restrict__ mono,
                    const float* __restrict__ warp,
                    const float* __restrict__ pprev,
                    const float* __restrict__ reach,
                    float* __restrict__ out) {
  __shared__ float smem[LDSF];

  const int row       = blockIdx.y;
  const int tileStart = blockIdx.x * TILE;
  const int b         = row >> 1;
  const float* src    = mono + (size_t)b * Tn;
  const int g0        = tileStart - HALO;

  // Block-uniform: gather never reads ahead (w <= 0, prefix from past chunks),
  // and never further back than reach[row] samples. Halo check is conservative (+8 slack).
  const bool useLds = (reach[row] + 8.0f) < (float)HALO;

  if (useLds) {
    const int nvec = (HALO + TILE) / 4;           // 2560 B128 transfers
    for (int i = threadIdx.x; i < nvec; i += blockDim.x) {
      int gidx = g0 + i * 4;                      // 16B-aligned (g0 % 4 == 0)
      if (gidx >= 0 && gidx < Tn)
        async_g2l_b128(src + gidx, &smem[i * 4]);
    }
    wait_async_zero();                            // own wave's ASYNCcnt -> 0
    __syncthreads();                              // whole tile visible in LDS
  }

  const float* wrow = warp  + (size_t)row * Kn;
  const float* prow = pprev + (size_t)row * Kn;
  float*       orow = out   + (size_t)row * Tn;

  for (int j = threadIdx.x * 4; j < TILE; j += blockDim.x * 4) {
    int t0 = tileStart + j;
    if (t0 >= Tn) continue;      // Tn % 4 == 0 and t0 % 4 == 0 -> full vec ok
    v4f r;
    #pragma unroll
    for (int c = 0; c < 4; ++c) {
      int t = t0 + c;
      int k = t / CHUNK;
      float w  = wrow[k];
      float pp = prow[k];
      float p = fminf(fmaxf(w + (float)t, 0.0f), (float)(Tn - 1));
      p = fmaxf(p, pp);               // == cummax(pos)(t)
      float lf = floorf(p);
      float a  = p - lf;
      int li = (int)lf;
      int ri = (int)ceilf(p);
      float gl, gr;
      if (useLds) { gl = smem[li - g0]; gr = smem[ri - g0]; }
      else        { gl = src[li];       gr = src[ri];       }
      r[c] = (1.0f - a) * gl + a * gr;
    }
    // Write-once streaming output: non-temporal B128 store, keep L2 for mono.
    __builtin_nontemporal_store(r, (v4f*)(orow + t0));
  }
}

// ---------------------------------------------------------------------------
// Host launch. Workspace layout (floats): warp[32*1200] | pprev[32*1200] | reach[32]
// = ~307 KB, well within scratch.
// ---------------------------------------------------------------------------
extern "C" void kernel_launch(void* const* d_in, const int* in_sizes, int n_in,
                              void* d_out, int out_size, void* d_ws, size_t ws_size,
                              hipStream_t stream) {
  const float* mono = (const float*)d_in[0];   // (16, 1, 480000) f32
  const float* view = (const float*)d_in[1];   // (16, 7, 1200)   f32
  float* out = (float*)d_out;                  // (16, 2, 480000) f32

  float* warp  = (float*)d_ws;
  float* pprev = warp  + (size_t)ROWS * Kn;
  float* reach = pprev + (size_t)ROWS * Kn;

  (void)in_sizes; (void)n_in; (void)out_size; (void)ws_size;

  int n1 = Bn * Kn;
  gw_disp_kernel<<<(n1 + 255) / 256, 256, 0, stream>>>(view, warp);
  gw_scan_kernel<<<ROWS, 32, 0, stream>>>(warp, pprev, reach);

  dim3 grid((Tn + TILE - 1) / TILE, ROWS);
  gw_warp_kernel<<<grid, 256, 0, stream>>>(mono, warp, pprev, reach, out);
}